// SSKernelDiag_31310311588114
// MI455X (gfx1250) — compile-verified
//
#include <hip/hip_runtime.h>
#include <hip/hip_bf16.h>
#include <math.h>

// S4D SSM convolution kernel K[h,l] for MI455X (gfx1250, wave32, WMMA).
//
// Factorization: l = 32a + b  =>  exp(dtA*l) = P[n,a]*Q[n,b]
//   K[h, 32a+b] = 2*Re( sum_n (CB*P)[a,n] * Q[n,b] )
//               = 2*( Ar @ Qr - Ai @ Qi )[a,b]
// Two real (32x64)@(64x32) GEMMs per head on v_wmma_f32_16x16x4_f32.

typedef __attribute__((ext_vector_type(2))) float v2f;
typedef __attribute__((ext_vector_type(8))) float v8f;

#define NSTATE 64
#define LLEN   1024   // = 32 * 32

__global__ __launch_bounds__(128)
void s4d_wmma_kernel(const float* __restrict__ log_dt,
                     const float* __restrict__ Lam_re,
                     const float* __restrict__ Lam_im,
                     const float* __restrict__ Bmat,
                     const float* __restrict__ C_re,
                     const float* __restrict__ C_im,
                     const float* __restrict__ Dvec,
                     float* __restrict__ out)
{
    const int h    = blockIdx.x;
    const int t    = threadIdx.x;       // 0..127
    const int lane = t & 31;
    const int wave = t >> 5;            // 0..3

    // Padded LDS to dodge bank conflicts on fragment gathers.
    __shared__ float sQr [NSTATE][33];  // Q[n][b]   (b = low 5 bits of l)
    __shared__ float sQi [NSTATE][33];
    __shared__ float sAr [32][65];      // (CB*P)[a][n] real
    __shared__ float sAiN[32][65];      // -(CB*P)[a][n] imag  (pre-negated)

    // ---------------- Phase 1: generate Vandermonde factors -------------
    {
        const int n   = t & 63;
        const int idx = h * NSTATE + n;
        const float dt  = expf(log_dt[h]);
        const float lre = Lam_re[idx];
        const float lim = Lam_im[idx];
        const float dre = dt * lre;     // Re(dtA) <= 0 (stable init)
        const float dim = dt * lim;

        if (t < 64) {
            // Q[n][b] = exp(dtA * b), b = 0..31
            for (int b = 0; b < 32; ++b) {
                float e = expf(dre * (float)b);
                float s, c;
                sincosf(dim * (float)b, &s, &c);
                sQr[n][b] = e * c;
                sQi[n][b] = e * s;
            }
        } else {
            // CB[n] = C * (expm1(dtA)/Lam) * B
            float e1 = expf(dre);
            float s1, c1;
            sincosf(dim, &s1, &c1);
            float em1r = e1 * c1 - 1.0f;            // Re expm1(dtA)
            float em1i = e1 * s1;                   // Im expm1(dtA)
            float inv  = 1.0f / (lre * lre + lim * lim);
            float bv   = Bmat[idx];
            float bbr  = (em1r * lre + em1i * lim) * inv * bv;   // B_bar
            float bbi  = (em1i * lre - em1r * lim) * inv * bv;
            float cr   = C_re[idx];
            float ci   = C_im[idx];
            float cbr  = cr * bbr - ci * bbi;       // CB
            float cbi  = cr * bbi + ci * bbr;
            // A[a][n] = CB * exp(dtA * 32a), a = 0..31  (imag pre-negated)
            for (int a = 0; a < 32; ++a) {
                float w = 32.0f * (float)a;
                float e = expf(dre * w);
                float s, c;
                sincosf(dim * w, &s, &c);
                float pr = e * c, pi = e * s;
                sAr [a][n] = cbr * pr - cbi * pi;
                sAiN[a][n] = -(cbr * pi + cbi * pr);
            }
        }
    }
    __syncthreads();

    // ---------------- Phase 2: two real GEMMs via WMMA -------------------
    // Wave w owns output tile (a0, b0); all lanes convergent (EXEC all-1s).
    const int a0 = (wave >> 1) * 16;
    const int b0 = (wave & 1) * 16;
    const int lr = lane & 15;           // low lane index
    const int lh = lane >> 4;           // lane half (0/1)

    v8f acc = {};
    for (int kk = 0; kk < 16; ++kk) {
        const int k0 = kk * 4;
        const int ka = k0 + lh * 2;     // this lane's K pair base
        v2f af, bf;

        // real part:  Ar @ Qr
        af.x = sAr[a0 + lr][ka + 0];
        af.y = sAr[a0 + lr][ka + 1];
        bf.x = sQr[ka + 0][b0 + lr];
        bf.y = sQr[ka + 1][b0 + lr];
        acc = __builtin_amdgcn_wmma_f32_16x16x4_f32(
                  false, af, false, bf, (short)0, acc, false, false);

        // imag part: (-Ai) @ Qi
        af.x = sAiN[a0 + lr][ka + 0];
        af.y = sAiN[a0 + lr][ka + 1];
        bf.x = sQi[ka + 0][b0 + lr];
        bf.y = sQi[ka + 1][b0 + lr];
        acc = __builtin_amdgcn_wmma_f32_16x16x4_f32(
                  false, af, false, bf, (short)0, acc, false, false);
    }

    // ---------------- Store: K = 2*acc (+ D[h] at l==0) ------------------
    float* outh = out + (size_t)h * LLEN;
    #pragma unroll
    for (int v = 0; v < 8; ++v) {
        const int row = a0 + lh * 8 + v;    // a
        const int col = b0 + lr;            // b
        const int l   = row * 32 + col;
        float val = 2.0f * acc[v];
        if (l == 0) val += Dvec[h];
        outh[l] = val;
    }
}

extern "C" void kernel_launch(void* const* d_in, const int* in_sizes, int n_in,
                              void* d_out, int out_size, void* d_ws, size_t ws_size,
                              hipStream_t stream) {
    (void)n_in; (void)d_ws; (void)ws_size; (void)out_size;
    const float* log_dt = (const float*)d_in[0];
    const float* lam_re = (const float*)d_in[1];
    const float* lam_im = (const float*)d_in[2];
    const float* Bm     = (const float*)d_in[3];
    const float* c_re   = (const float*)d_in[4];
    const float* c_im   = (const float*)d_in[5];
    const float* Dv     = (const float*)d_in[6];
    // d_in[7] is L (python scalar); layout hardcodes L=1024=32*32, N=64.
    const int H = in_sizes[0];

    s4d_wmma_kernel<<<dim3(H), dim3(128), 0, stream>>>(
        log_dt, lam_re, lam_im, Bm, c_re, c_im, Dv, (float*)d_out);
}